// GCN_blk_687194767617
// MI455X (gfx1250) — compile-verified
//
#include <hip/hip_runtime.h>

// GCN (2-layer, symmetric-norm, self-loops) for MI455X / gfx1250.
// Strategy: edge list (102 MB) stays resident in the 192 MB L2 after first
// pass; node tables (~3 MB each) are L2-resident, so gathers + f32 atomics
// are L2-local. Node transforms use V_WMMA_F32_16X16X4_F32 (wave32, 16 nodes
// per wave). N = 100000 = 6250 * 16 exactly -> no partial WMMA tiles.

typedef __attribute__((ext_vector_type(2))) float v2f;
typedef __attribute__((ext_vector_type(8))) float v8f;

#define THREADS 256

// ---------------------------------------------------------------- init / degree
__global__ void k_init_deg(unsigned* __restrict__ deg, int n) {
  int i = blockIdx.x * blockDim.x + threadIdx.x;
  if (i < n) deg[i] = 1u;  // self-loop pre-counted
}

__global__ void k_degree(const long long* __restrict__ dst,
                         unsigned* __restrict__ deg, int e) {
  int i = blockIdx.x * blockDim.x + threadIdx.x;
  if (i >= e) return;
  __builtin_prefetch(dst + i + 4096, 0, 1);  // global_prefetch_b8 (speculative)
  atomicAdd(&deg[(int)dst[i]], 1u);
}

__global__ void k_dinv(const unsigned* __restrict__ deg,
                       float* __restrict__ dinv, int n) {
  int i = blockIdx.x * blockDim.x + threadIdx.x;
  if (i >= n) return;
  float d = (float)deg[i];
  dinv[i] = (d > 0.f) ? rsqrtf(d) : 0.f;
}

// ------------------------------------------------- layer 1 node transform (WMMA)
// hs1[n][c] = (x[n] @ W1)[c] * dinv[n],  agg1 initialized to hs1 (self-loop term)
// A = x tile 16x5 (K padded to 8 via two 16x16x4 WMMAs), B = W1 (5x8 in 16 cols)
__global__ void k_layer1(const float* __restrict__ x,
                         const float* __restrict__ W1,
                         const float* __restrict__ dinv,
                         float* __restrict__ hs1,
                         float* __restrict__ agg1, int nwaves) {
  const int wave = blockIdx.x * (blockDim.x >> 5) + (threadIdx.x >> 5);
  if (wave >= nwaves) return;  // wave-uniform: surviving waves have EXEC all-1s
  const int lane = threadIdx.x & 31;
  const int col  = lane & 15;   // D column (valid < 8)
  const int half = lane >> 4;
  const int base = wave * 16;
  const int row  = base + col;  // A row for this lane

  // A fragment (ISA 7.12.2: V0 = K0|K2, V1 = K1|K3 split by lane half)
  const float* xr = x + (size_t)row * 5;
  v2f a0, a1;
  a0.x = xr[half ? 2 : 0];
  a0.y = xr[half ? 3 : 1];
  a1.x = half ? 0.f : xr[4];   // K=4 (K=5..7 padded with zero)
  a1.y = 0.f;

  // B fragment: W1 is [5][8] row-major; columns 8..15 zero-padded
  v2f b0 = {0.f, 0.f}, b1v = {0.f, 0.f};
  if (col < 8) {
    b0.x  = W1[(half ? 2 : 0) * 8 + col];
    b0.y  = W1[(half ? 3 : 1) * 8 + col];
    b1v.x = half ? 0.f : W1[4 * 8 + col];
  }

  v8f c = {};
  c = __builtin_amdgcn_wmma_f32_16x16x4_f32(false, a0, false, b0, (short)0, c, false, false);
  c = __builtin_amdgcn_wmma_f32_16x16x4_f32(false, a1, false, b1v, (short)0, c, false, false);

  // D layout: lane (half, col) holds rows M = half*8 + r in c[r]
  if (col < 8) {
#pragma unroll
    for (int r = 0; r < 8; ++r) {
      int node = base + half * 8 + r;
      float hv = c[r] * dinv[node];
      hs1[(size_t)node * 8 + col]  = hv;
      agg1[(size_t)node * 8 + col] = hv;
    }
  }
}

// -------------------------------------------------------- edge scatter (atomics)
template <int NC>
__global__ void k_edge(const long long* __restrict__ src,
                       const long long* __restrict__ dst,
                       const float* __restrict__ hs,
                       float* __restrict__ agg, int e) {
  int i = blockIdx.x * blockDim.x + threadIdx.x;
  if (i >= e) return;
  __builtin_prefetch(src + i + 4096, 0, 1);
  __builtin_prefetch(dst + i + 4096, 0, 1);
  int s = (int)src[i];
  int d = (int)dst[i];
  const float* hp = hs + (size_t)s * 8;   // stride-8, 32B aligned
  float* ap = agg + (size_t)d * 8;
  float4 h0 = *(const float4*)hp;
  atomicAdd(ap + 0, h0.x);
  atomicAdd(ap + 1, h0.y);
  atomicAdd(ap + 2, h0.z);
  atomicAdd(ap + 3, h0.w);
  if constexpr (NC == 8) {
    float4 h1 = *(const float4*)(hp + 4);
    atomicAdd(ap + 4, h1.x);
    atomicAdd(ap + 5, h1.y);
    atomicAdd(ap + 6, h1.z);
    atomicAdd(ap + 7, h1.w);
  } else {  // NC == 5
    atomicAdd(ap + 4, hp[4]);
  }
}

// ------------------------------------------------- layer 2 node transform (WMMA)
// t[n]   = agg1[n]*dinv[n] + b1          (layer-1 output, 8 ch)
// hs2[n] = (t[n] @ W2) * dinv[n]         (5 ch, stored stride 8, padded 0)
__global__ void k_layer2(const float* __restrict__ agg1,
                         const float* __restrict__ dinv,
                         const float* __restrict__ bias1,
                         const float* __restrict__ W2,
                         float* __restrict__ hs2,
                         float* __restrict__ agg2, int nwaves) {
  const int wave = blockIdx.x * (blockDim.x >> 5) + (threadIdx.x >> 5);
  if (wave >= nwaves) return;
  const int lane = threadIdx.x & 31;
  const int col  = lane & 15;   // D column (valid < 5)
  const int half = lane >> 4;
  const int base = wave * 16;
  const int row  = base + col;

  float din = dinv[row];
  const float* ar = agg1 + (size_t)row * 8;
  const int k0 = half * 2;
  // A fragments: wmma#1 covers K=0..3, wmma#2 covers K=4..7
  v2f a0, a1;
  a0.x = ar[k0]     * din + bias1[k0];
  a0.y = ar[k0 + 1] * din + bias1[k0 + 1];
  a1.x = ar[4 + k0]     * din + bias1[4 + k0];
  a1.y = ar[4 + k0 + 1] * din + bias1[4 + k0 + 1];

  // B fragments: W2 is [8][5] row-major; columns 5..15 zero-padded
  v2f b0 = {0.f, 0.f}, b1f = {0.f, 0.f};
  if (col < 5) {
    b0.x  = W2[(k0)     * 5 + col];
    b0.y  = W2[(k0 + 1) * 5 + col];
    b1f.x = W2[(4 + k0)     * 5 + col];
    b1f.y = W2[(4 + k0 + 1) * 5 + col];
  }

  v8f c = {};
  c = __builtin_amdgcn_wmma_f32_16x16x4_f32(false, a0, false, b0, (short)0, c, false, false);
  c = __builtin_amdgcn_wmma_f32_16x16x4_f32(false, a1, false, b1f, (short)0, c, false, false);

  if (col < 8) {
#pragma unroll
    for (int r = 0; r < 8; ++r) {
      int node = base + half * 8 + r;
      float hv = (col < 5) ? c[r] * dinv[node] : 0.f;  // pad cols 5..7 to zero
      hs2[(size_t)node * 8 + col]  = hv;
      agg2[(size_t)node * 8 + col] = hv;
    }
  }
}

// ------------------------------------------------------------------ final scale
__global__ void k_finish(const float* __restrict__ agg2,
                         const float* __restrict__ dinv,
                         const float* __restrict__ bias2,
                         float* __restrict__ out, int n) {
  int i = blockIdx.x * blockDim.x + threadIdx.x;
  if (i >= n) return;
  float din = dinv[i];
#pragma unroll
  for (int c = 0; c < 5; ++c)
    out[(size_t)i * 5 + c] = agg2[(size_t)i * 8 + c] * din + bias2[c];
}

// ----------------------------------------------------------------------- launch
static inline size_t align256(size_t v) { return (v + 255) & ~(size_t)255; }

extern "C" void kernel_launch(void* const* d_in, const int* in_sizes, int n_in,
                              void* d_out, int out_size, void* d_ws, size_t ws_size,
                              hipStream_t stream) {
  const float*     x    = (const float*)d_in[0];       // [N,5]
  const long long* eidx = (const long long*)d_in[1];   // [2,E] int64
  // d_in[2] (edge_f), d_in[3] (edge_attr) unused by reference
  const float* W1 = (const float*)d_in[4];             // [5,8]
  const float* b1 = (const float*)d_in[5];             // [8]
  const float* W2 = (const float*)d_in[6];             // [8,5]
  const float* b2 = (const float*)d_in[7];             // [5]
  float* out = (float*)d_out;

  const int N = in_sizes[0] / 5;
  const int E = in_sizes[1] / 2;
  const long long* srcp = eidx;
  const long long* dstp = eidx + E;

  // workspace carve-out (256B aligned blocks)
  char* ws = (char*)d_ws;
  size_t o = 0;
  unsigned* deg  = (unsigned*)(ws + o); o += align256((size_t)N * 4);
  float*    dinv = (float*)(ws + o);    o += align256((size_t)N * 4);
  float*    hs1  = (float*)(ws + o);    o += align256((size_t)N * 8 * 4);
  float*    agg1 = (float*)(ws + o);    o += align256((size_t)N * 8 * 4);
  float*    hs2  = (float*)(ws + o);    o += align256((size_t)N * 8 * 4);
  float*    agg2 = (float*)(ws + o);    o += align256((size_t)N * 8 * 4);
  (void)ws_size; (void)n_in; (void)out_size;

  const int nwaves  = N / 16;                       // N=100000 -> 6250 exact
  const int wblocks = (nwaves + 7) / 8;             // 8 waves per 256-thr block
  const int nblk = (N + THREADS - 1) / THREADS;
  const int eblk = (E + THREADS - 1) / THREADS;

  k_init_deg<<<nblk, THREADS, 0, stream>>>(deg, N);
  k_degree<<<eblk, THREADS, 0, stream>>>(dstp, deg, E);
  k_dinv<<<nblk, THREADS, 0, stream>>>(deg, dinv, N);

  k_layer1<<<wblocks, THREADS, 0, stream>>>(x, W1, dinv, hs1, agg1, nwaves);
  k_edge<8><<<eblk, THREADS, 0, stream>>>(srcp, dstp, hs1, agg1, E);

  k_layer2<<<wblocks, THREADS, 0, stream>>>(agg1, dinv, b1, W2, hs2, agg2, nwaves);
  k_edge<5><<<eblk, THREADS, 0, stream>>>(srcp, dstp, hs2, agg2, E);

  k_finish<<<nblk, THREADS, 0, stream>>>(agg2, dinv, b2, out, N);
}